// SparseGCNLayer_30777735643318
// MI455X (gfx1250) — compile-verified
//
#include <hip/hip_runtime.h>
#include <hip/hip_bf16.h>

// CDNA5 / gfx1250. wave32. fp32 WMMA (16x16x4) for the dense projection.

typedef __attribute__((ext_vector_type(2))) float v2f;
typedef __attribute__((ext_vector_type(8))) float v8f;

#define C_DIM 128  // C_IN == C_OUT == 128 in the reference

// ---------------------------------------------------------------------------
// Kernel 1: grid-stride zero fill (deg and out must start at 0; harness
// poisons d_out/d_ws with 0xAA).
// ---------------------------------------------------------------------------
__global__ void gcn_zero_kernel(float* __restrict__ p, long n) {
    long i = (long)blockIdx.x * blockDim.x + threadIdx.x;
    const long stride = (long)gridDim.x * blockDim.x;
    for (; i < n; i += stride) p[i] = 0.0f;
}

// ---------------------------------------------------------------------------
// Kernel 2: degree accumulation. deg[rows[e]] += vals[e]
// ---------------------------------------------------------------------------
__global__ void gcn_deg_kernel(const int* __restrict__ rows,
                               const float* __restrict__ vals,
                               float* __restrict__ deg, int E) {
    int e = blockIdx.x * blockDim.x + threadIdx.x;
    if (e < E) atomicAdd(deg + rows[e], vals[e]);
}

// ---------------------------------------------------------------------------
// Kernel 3: projection  H[n][o] = sum_c X[n][c] * W[o][c] + bias[o]
// using V_WMMA_F32_16X16X4_F32.
//
// Block = 256 threads = 8 waves. Block b owns rows [16b, 16b+16); wave w owns
// output columns [16w, 16w+16). Each wave chains 32 WMMAs over K=128.
//
// Fragment layouts (ISA 7.12.2, wave32):
//   A 16x4 f32 : lane l holds M = l&15, K = {off, off+1}, off = (l>>4)*2
//   B 4x16 f32 : lane l holds N = l&15, K = {off, off+1}   (B[k][o] = W[o][k])
//   C/D 16x16  : lane l, VGPR v holds row (v + (l>>4)*8), col l&15
// ---------------------------------------------------------------------------
__global__ __launch_bounds__(256) void gcn_gemm_wmma_kernel(
    const float* __restrict__ X, const float* __restrict__ W,
    const float* __restrict__ bias, float* __restrict__ H, int N) {
    const int lane  = threadIdx.x & 31;
    const int wave  = threadIdx.x >> 5;     // o-tile index, 0..7
    const int m     = lane & 15;
    const int off   = (lane >> 4) << 1;     // 0 or 2

    int xr = blockIdx.x * 16 + m;           // A-fragment source row (node)
    if (xr >= N) xr = N - 1;                // clamp (N is a multiple of 16 here)
    const long xrow = (long)xr * C_DIM;
    const long wrow = (long)(wave * 16 + m) * C_DIM;  // B col -> W row (o)

    v8f acc = {};
#pragma unroll
    for (int kc = 0; kc < C_DIM; kc += 4) {
        v2f a = *(const v2f*)(X + xrow + kc + off);
        v2f b = *(const v2f*)(W + wrow + kc + off);
        // (neg_a, A, neg_b, B, c_mod, C, reuse_a, reuse_b)
        acc = __builtin_amdgcn_wmma_f32_16x16x4_f32(
            false, a, false, b, (short)0, acc, false, false);
    }

    const float bv      = bias[wave * 16 + m];
    const int   colbase = wave * 16 + m;
    const int   rowbase = blockIdx.x * 16 + ((lane >> 4) << 3);
#pragma unroll
    for (int v = 0; v < 8; ++v) {
        const int r = rowbase + v;
        if (r < N) H[(long)r * C_DIM + colbase] = acc[v] + bv;
    }
}

// ---------------------------------------------------------------------------
// Kernel 4: fused SpMM + degree normalization.
// One wave per edge; lane handles 4 channels (float4 gather from L2-resident
// H, 4x global_atomic_add_f32 scatter). scale = vals[e]/deg[rows[e]] fuses
// the final division.
// ---------------------------------------------------------------------------
__global__ __launch_bounds__(256) void gcn_spmm_kernel(
    const int* __restrict__ rows, const int* __restrict__ cols,
    const float* __restrict__ vals, const float* __restrict__ deg,
    const float* __restrict__ H, float* __restrict__ out, int E) {
    const int e    = (int)(((long)blockIdx.x * blockDim.x + threadIdx.x) >> 5);
    const int lane = threadIdx.x & 31;
    if (e >= E) return;

    const int   r     = rows[e];
    const int   c     = cols[e];
    const float scale = vals[e] / deg[r];

    const float4 hv = *(const float4*)(H + (long)c * C_DIM + lane * 4);
    float* op = out + (long)r * C_DIM + lane * 4;
    atomicAdd(op + 0, hv.x * scale);
    atomicAdd(op + 1, hv.y * scale);
    atomicAdd(op + 2, hv.z * scale);
    atomicAdd(op + 3, hv.w * scale);
}

// ---------------------------------------------------------------------------
// Host launcher. Inputs: 0=node_feats[N,128] f32, 1=rows[E] i32, 2=cols[E] i32,
// 3=vals[E] f32, 4=W[128,128] f32, 5=b[128] f32. Output: [N,128] f32.
// Workspace layout: deg[N] | H[N*128]
// ---------------------------------------------------------------------------
extern "C" void kernel_launch(void* const* d_in, const int* in_sizes, int n_in,
                              void* d_out, int out_size, void* d_ws, size_t ws_size,
                              hipStream_t stream) {
    const float* X    = (const float*)d_in[0];
    const int*   rows = (const int*)d_in[1];
    const int*   cols = (const int*)d_in[2];
    const float* vals = (const float*)d_in[3];
    const float* W    = (const float*)d_in[4];
    const float* bias = (const float*)d_in[5];
    float*       out  = (float*)d_out;

    const int N = in_sizes[0] / C_DIM;   // 50000
    const int E = in_sizes[1];           // 850000

    float* deg = (float*)d_ws;
    float* H   = deg + ((N + 3) & ~3);   // 16B-aligned for float4 gathers

    // 1) zero deg and out
    {
        int blocks = (N + 255) / 256;
        if (blocks > 2048) blocks = 2048;
        gcn_zero_kernel<<<blocks, 256, 0, stream>>>(deg, (long)N);
        long n_out = (long)N * C_DIM;
        int blocks2 = (int)((n_out + 255) / 256);
        if (blocks2 > 8192) blocks2 = 8192;
        gcn_zero_kernel<<<blocks2, 256, 0, stream>>>(out, n_out);
    }

    // 2) degree
    gcn_deg_kernel<<<(E + 255) / 256, 256, 0, stream>>>(rows, vals, deg, E);

    // 3) WMMA projection
    gcn_gemm_wmma_kernel<<<(N + 15) / 16, 256, 0, stream>>>(X, W, bias, H, N);

    // 4) fused SpMM + normalize (one wave32 per edge)
    {
        long waves  = (long)E;
        long blocks = (waves * 32 + 255) / 256;
        gcn_spmm_kernel<<<(int)blocks, 256, 0, stream>>>(rows, cols, vals, deg, H, out, E);
    }
}